// HighPassFilter_84061099918039
// MI455X (gfx1250) — compile-verified
//
#include <hip/hip_runtime.h>

typedef __attribute__((ext_vector_type(2))) float v2f;
typedef __attribute__((ext_vector_type(8))) float v8f;

// LDS geometry (floats). Strides padded for bank-conflict-free B fetches:
//  - stage-1 B: two half-waves read rows j and j+2 at 16 consecutive cols;
//    2*IN_STRIDE % 64 == 16 -> the two halves hit disjoint bank groups.
//  - stage-2 B: 16 lanes read 16 different rows at one col;
//    MID_STRIDE % 64 == 17 (odd, coprime with 64) -> all distinct banks.
#define IN_ROWS    44      // 42 real halo rows + 2 zero K-pad rows
#define IN_STRIDE  264     // 256 data cols + zero pad
#define MID_ROWS   16
#define MID_STRIDE 273     // col m stored at (m + 8); m in [-8, 264]

// D = A x B + C, fp32 16x16x4 (11 K-chunks cover the 44-wide banded filter).
// A (16x4) layout per ISA: lane<16 -> M=lane, {V0,V1}={K0,K1}; lane>=16 -> {K2,K3}.
// B (4x16) symmetric: V0 = row K0 (lanes 0-15) / row K2 (lanes 16-31), V1 = K1/K3.
// C/D: VGPR v, lane l -> M = v + 8*(l>>4), N = l&15.
__launch_bounds__(256)
__global__ void upfirdn2d_sym6_wmma(const float* __restrict__ in,
                                    const float* __restrict__ k2d,
                                    float* __restrict__ out) {
    __shared__ float s_in[IN_ROWS * IN_STRIDE];      // 46464 B
    __shared__ float s_mid[MID_ROWS * MID_STRIDE];   // 17472 B
    __shared__ float s_h[12];                        // reversed 1-D taps

    const int tid  = threadIdx.x;
    const int lane = tid & 31;
    // Wave index is uniform across the wave: force it scalar so tile addressing
    // stays on the SALU and no divergent control flow is generated around it.
    const int wave = __builtin_amdgcn_readfirstlane(tid >> 5);

    const int img = blockIdx.x;          // image index (B*C folded)
    const int R0  = blockIdx.y << 4;     // first output row of this strip

    const float* __restrict__ src = in  + (size_t)img * 256u * 256u;
    float*       __restrict__ dst = out + (size_t)img * 128u * 128u;

    // Recover reversed 1-D taps from the separable 2-D kernel: h[i]=K[i,6]/sqrt(K[6,6]).
    // Reference = correlation with flipped kernel => out[r] = sum_k H[k] in[2r+k-5], H[k]=h[11-k].
    if (tid < 12) {
        float s = rsqrtf(k2d[6 * 12 + 6]);
        s_h[tid] = k2d[(11 - tid) * 12 + 6] * s;
    }

    // Stage-0: cooperative halo load. Thread t owns input column t (fully coalesced
    // 1 KB bursts per row). Rows j=0..41 map to input row 2*R0 + j - 5 (zero-clamped);
    // rows 42,43 are K-padding. Also zero LDS col pad and the mid border columns.
    {
        const int c = tid;                       // 0..255
        #pragma unroll 4
        for (int j = 0; j < IN_ROWS; ++j) {
            const int r = 2 * R0 + j - 5;
            float v = 0.0f;
            if (j < 42 && r >= 0 && r < 256) v = src[r * 256 + c];
            s_in[j * IN_STRIDE + c] = v;
            if (c < IN_STRIDE - 256) s_in[j * IN_STRIDE + 256 + c] = 0.0f;
        }
        if (tid < 17) {                          // mid cols m=-8..-1 and m=256..264
            const int pc = (tid < 8) ? tid : (264 + (tid - 8));
            #pragma unroll
            for (int r = 0; r < MID_ROWS; ++r) s_mid[r * MID_STRIDE + pc] = 0.0f;
        }
    }
    __syncthreads();

    // Banded filter operand, identical for both passes: A[i,j] = H[j - 2i], j in [0,44).
    v2f A[11];
    {
        const int m  = lane & 15;
        const int ho = lane >> 4;
        #pragma unroll
        for (int q = 0; q < 11; ++q) {
            const int j0 = 4 * q + 2 * ho;
            const int k0 = j0 - 2 * m;
            const int k1 = k0 + 1;
            v2f a;
            a.x = (k0 >= 0 && k0 < 12) ? s_h[k0] : 0.0f;
            a.y = (k1 >= 0 && k1 < 12) ? s_h[k1] : 0.0f;
            A[q] = a;
        }
    }

    const int n  = lane & 15;
    const int hi = lane >> 4;

    // Stage 1 (vertical, WMMA): Mid[16 x 256] = A[16x44] x In[44 x 256].
    // 16 col-tiles; wave w owns tiles w and w+8 (statically unrolled: no
    // divergent loop, trip count is exactly 2).
    #pragma unroll
    for (int it = 0; it < 2; ++it) {
        const int c0 = (wave + 8 * it) << 4;
        v8f acc = {};
        #pragma unroll
        for (int q = 0; q < 11; ++q) {
            const int j = 4 * q + 2 * hi;
            v2f b;
            b.x = s_in[j * IN_STRIDE + c0 + n];
            b.y = s_in[(j + 1) * IN_STRIDE + c0 + n];
            acc = __builtin_amdgcn_wmma_f32_16x16x4_f32(
                false, A[q], false, b, (short)0, acc, false, false);
        }
        #pragma unroll
        for (int v = 0; v < 8; ++v)              // D: M = v+8*hi (mid row), N = col
            s_mid[(v + 8 * hi) * MID_STRIDE + 8 + c0 + n] = acc[v];
    }
    __syncthreads();

    // Stage 2 (horizontal, WMMA, transposed): Out^T[16x16] = A x Mid^T[44x16].
    // B[j, r] = mid[r, 2*C0 + j - 5]  (LDS col = 2*C0 + 3 + j; max 270 < 273).
    {
        const int C0 = wave << 4;
        v8f acc = {};
        #pragma unroll
        for (int q = 0; q < 11; ++q) {
            const int j = 4 * q + 2 * hi;
            const float* row = &s_mid[n * MID_STRIDE + 2 * C0 + 3 + j];
            v2f b;
            b.x = row[0];
            b.y = row[1];
            acc = __builtin_amdgcn_wmma_f32_16x16x4_f32(
                false, A[q], false, b, (short)0, acc, false, false);
        }
        // D holds Out^T: M = out col offset, N = out row offset.
        #pragma unroll
        for (int v = 0; v < 8; ++v)
            dst[(R0 + n) * 128 + C0 + v + 8 * hi] = acc[v];
    }
}

extern "C" void kernel_launch(void* const* d_in, const int* in_sizes, int n_in,
                              void* d_out, int out_size, void* d_ws, size_t ws_size,
                              hipStream_t stream) {
    (void)n_in; (void)out_size; (void)d_ws; (void)ws_size;
    const float* x   = (const float*)d_in[0];   // [16,64,256,256] fp32
    const float* k2d = (const float*)d_in[1];   // [12,12] fp32 (h outer h)
    float* y = (float*)d_out;                   // [16,64,128,128] fp32

    const int images = in_sizes[0] >> 16;       // (B*C) = N / (256*256)
    dim3 grid((unsigned)images, 8u, 1u);        // 8 strips of 16 output rows
    dim3 block(256u, 1u, 1u);                   // 8 wave32s
    upfirdn2d_sym6_wmma<<<grid, block, 0, stream>>>(x, k2d, y);
}